// KVCache_16303695855978
// MI455X (gfx1250) — compile-verified
//
#include <hip/hip_runtime.h>

// Problem constants from the reference (fixed shapes).
#define B_ 8
#define H_ 16
#define T_ 16
#define D_ 128
#define S_ 4096

typedef float v4f __attribute__((ext_vector_type(4)));
// Exact pointee type the gfx1250 async builtins expect (per hipcc diagnostic):
// "__attribute__((__vector_size__(4 * sizeof(int)))) int"
typedef int v4i_vs __attribute__((vector_size(4 * sizeof(int))));

#if defined(__has_builtin)
# if __has_builtin(__builtin_amdgcn_global_load_async_to_lds_b128) && \
     __has_builtin(__builtin_amdgcn_global_store_async_from_lds_b128) && \
     __has_builtin(__builtin_amdgcn_s_wait_asynccnt)
#  define USE_ASYNC_LDS 1
# endif
#endif

// ---------------------------------------------------------------------------
// Bulk streaming copy: cache -> out, non-temporal both ways (536 MB stream,
// 2.8x the 192 MB L2, so NT retention is the right policy). Each block moves
// a 32 KB tile: 256 threads x 8 float4 (b128) NT loads clustered for MLP,
// then 8 NT b128 stores. NO bounds checks: the launch guarantees exact
// tiling (n4 % 2048 == 0 for these shapes); a guarded tail kernel covers
// any remainder. 32-bit indexing keeps addressing to one base computation
// plus immediate instruction offsets (u*4096 bytes fits the 24-bit ioffset).
// ---------------------------------------------------------------------------
#define CP_UNROLL 8
#define CP_TILE   (256 * CP_UNROLL)  // float4 per block

__global__ __launch_bounds__(256) void kv_copy_nt(const v4f* __restrict__ s4,
                                                  v4f* __restrict__ d4) {
    const int base = (int)blockIdx.x * CP_TILE + (int)threadIdx.x;

    v4f r[CP_UNROLL];
#pragma unroll
    for (int u = 0; u < CP_UNROLL; ++u)
        r[u] = __builtin_nontemporal_load(s4 + (base + u * 256));
#pragma unroll
    for (int u = 0; u < CP_UNROLL; ++u)
        __builtin_nontemporal_store(r[u], d4 + (base + u * 256));
}

// Tail: guarded scalar float4 copy for any non-tile remainder (not expected
// for the reference shapes; launched only when rem != 0).
__global__ __launch_bounds__(256) void kv_copy_tail(const v4f* __restrict__ s4,
                                                    v4f* __restrict__ d4,
                                                    int start, int n4) {
    const int i = start + (int)(blockIdx.x * blockDim.x + threadIdx.x);
    if (i < n4) __builtin_nontemporal_store(__builtin_nontemporal_load(s4 + i), d4 + i);
}

// ---------------------------------------------------------------------------
// Scatter: one wave32 per (b,h,t) row (D=128 fp32 = 32 lanes x b128).
// waves [0, 2048) handle k, [2048, 4096) handle v. Row goes to
// out[tensor][b][h][input_pos[b][t]][:]. Uses the gfx1250 async
// global<->LDS path (ASYNCcnt) when available, else direct b128 copy.
// This stage is ~0.4% of total traffic, so the wait_asynccnt(0) drain
// between load and store is immaterial.
// ---------------------------------------------------------------------------
__global__ __launch_bounds__(256) void kv_scatter(const int* __restrict__ pos,
                                                  const float* __restrict__ k,
                                                  const float* __restrict__ v,
                                                  float* __restrict__ out,
                                                  long long cacheElems) {
    const int wid  = (int)((blockIdx.x * blockDim.x + threadIdx.x) >> 5);
    const int lane = (int)(threadIdx.x & 31);

    const int NR = B_ * H_ * T_;  // 2048 rows per tensor
    int r = wid;
    const float* src;
    float*       dstBase;
    if (r < NR) {
        src = k;
        dstBase = out;
    } else {
        r -= NR;
        src = v;
        dstBase = out + cacheElems;
    }

    const int b   = r / (H_ * T_);
    const int rem = r % (H_ * T_);
    const int h   = rem / T_;
    const int t   = rem % T_;
    const int p   = pos[b * T_ + t];  // in [0, S)

    const v4f* s4 = (const v4f*)(src + (long long)((b * H_ + h) * T_ + t) * D_);
    v4f*       d4 = (v4f*)(dstBase + ((long long)(b * H_ + h) * S_ + (long long)p) * D_);

#if defined(USE_ASYNC_LDS)
    __shared__ v4f buf[256];  // one b128 slot per thread; waves use disjoint slots
    v4i_vs* lslot = (v4i_vs*)&buf[threadIdx.x];
    // memory -> LDS (async, per-lane b128)
    __builtin_amdgcn_global_load_async_to_lds_b128((v4i_vs*)(s4 + lane), lslot, 0, 0);
    __builtin_amdgcn_s_wait_asynccnt(0);
    // LDS -> memory (async, per-lane b128)
    __builtin_amdgcn_global_store_async_from_lds_b128((v4i_vs*)(d4 + lane), lslot, 0, 0);
    __builtin_amdgcn_s_wait_asynccnt(0);
#else
    d4[lane] = s4[lane];
#endif
}

// ---------------------------------------------------------------------------
// Launch: copies first, scatter second; same-stream ordering gives the
// copy -> scatter dependency. All graph-capture safe (no sync APIs).
// Inputs (setup_inputs order): 0=input_pos(i32), 1=k, 2=v, 3=k_cache, 4=v_cache.
// d_out = [k_out | v_out] flat.
// ---------------------------------------------------------------------------
extern "C" void kernel_launch(void* const* d_in, const int* in_sizes, int n_in,
                              void* d_out, int out_size, void* d_ws, size_t ws_size,
                              hipStream_t stream) {
    const int*   pos = (const int*)d_in[0];
    const float* k   = (const float*)d_in[1];
    const float* v   = (const float*)d_in[2];
    const float* kc  = (const float*)d_in[3];
    const float* vc  = (const float*)d_in[4];
    float*       out = (float*)d_out;

    const long long nCache = (long long)in_sizes[3];  // B*H*S*D = 67,108,864
    const int n4         = (int)(nCache / 4);         // 16,777,216 float4
    const int copyBlocks = n4 / CP_TILE;              // 8192 (exact for ref shapes)
    const int rem        = n4 - copyBlocks * CP_TILE; // 0 for ref shapes

    if (copyBlocks > 0) {
        kv_copy_nt<<<copyBlocks, 256, 0, stream>>>((const v4f*)kc, (v4f*)out);
        kv_copy_nt<<<copyBlocks, 256, 0, stream>>>((const v4f*)vc, (v4f*)(out + nCache));
    }
    if (rem > 0) {
        const int start = copyBlocks * CP_TILE;
        const int tb    = (rem + 255) / 256;
        kv_copy_tail<<<tb, 256, 0, stream>>>((const v4f*)kc, (v4f*)out, start, n4);
        kv_copy_tail<<<tb, 256, 0, stream>>>((const v4f*)vc, (v4f*)(out + nCache), start, n4);
    }

    const int totalWaves = 2 * B_ * H_ * T_;          // 4096 waves
    kv_scatter<<<totalWaves / 8, 256, 0, stream>>>(pos, k, v, out, nCache);
}